// GCN_21827023798811
// MI455X (gfx1250) — compile-verified
//
#include <hip/hip_runtime.h>
#include <math.h>

typedef __attribute__((ext_vector_type(2))) float v2f;
typedef __attribute__((ext_vector_type(8))) float v8f;

#define DIM 128

// ---------------------------------------------------------------------------
// copy x [n,DIM] into column block 0 of hcat [n, ldh]
__global__ void k_copy_x(const float* __restrict__ x, float* __restrict__ hcat,
                         int n, int ldh) {
  int g = blockIdx.x * blockDim.x + threadIdx.x;
  if (g >= n * DIM) return;
  int i = g >> 7, c = g & (DIM - 1);
  hcat[(size_t)i * ldh + c] = x[g];
}

// deg[i] = 1 (self loop)
__global__ void k_deg_init(float* __restrict__ deg, int n) {
  int i = blockIdx.x * blockDim.x + threadIdx.x;
  if (i < n) deg[i] = 1.0f;
}

// deg[dst[e]] += 1 (hardware f32 atomic, fire-and-forget)
__global__ void k_deg_edges(const long long* __restrict__ dst,
                            float* __restrict__ deg, int e) {
  int i = blockIdx.x * blockDim.x + threadIdx.x;
  if (i >= e) return;
  int d = (int)dst[i];
  __hip_atomic_fetch_add(&deg[d], 1.0f, __ATOMIC_RELAXED, __HIP_MEMORY_SCOPE_AGENT);
}

__global__ void k_dinv(const float* __restrict__ deg, float* __restrict__ dinv, int n) {
  int i = blockIdx.x * blockDim.x + threadIdx.x;
  if (i < n) {
    float d = deg[i];
    dinv[i] = (d > 0.0f) ? rsqrtf(d) : 0.0f;
  }
}

// ---------------------------------------------------------------------------
// C[m,o] = sum_k A[m,k] * B[o,k]  (+ bias[o])
// One wave = one 16x16 tile, accumulated with V_WMMA_F32_16X16X4_F32.
// Block = 256 threads = 8 waves covering the 8 column tiles (128 output cols).
// ISA layouts (32-bit, wave32):
//   A 16x4: lane&15 -> M row; VGPR v holds K = v + 2*(lane>>4)
//   B 4x16: lane&15 -> N col; VGPR v holds K = v + 2*(lane>>4)
//   C 16x16: VGPR j, lanes 0-15 -> M=j, lanes 16-31 -> M=j+8; N = lane&15
__global__ void k_gemm_wmma(const float* __restrict__ A, int lda,
                            const float* __restrict__ B, int ldb,
                            const float* __restrict__ bias,
                            float* __restrict__ C, int ldc,
                            int K, int nrows) {
  const int warp = threadIdx.x >> 5;        // 0..7 -> output column tile
  const int lane = threadIdx.x & 31;
  const int l15  = lane & 15;
  const int hi2  = (lane >> 4) << 1;        // 0 or 2 (K sub-offset)
  const int m0   = blockIdx.x << 4;
  const int o0   = warp << 4;

  int ra = m0 + l15;
  if (ra >= nrows) ra = nrows - 1;          // clamp (stores are guarded)
  const float* ap = A + (size_t)ra * lda + hi2;
  const float* bp = B + (size_t)(o0 + l15) * ldb + hi2;

  v8f c = {0.f, 0.f, 0.f, 0.f, 0.f, 0.f, 0.f, 0.f};
#pragma unroll 8
  for (int k = 0; k < K; k += 4) {
    v2f a = *(const v2f*)(ap + k);
    v2f b = *(const v2f*)(bp + k);
    c = __builtin_amdgcn_wmma_f32_16x16x4_f32(
        /*neg_a=*/false, a, /*neg_b=*/false, b,
        /*c_mod=*/(short)0, c, /*reuse_a=*/false, /*reuse_b=*/false);
  }

  const int col = o0 + l15;
  const float bv = bias ? bias[col] : 0.0f;
  const int row0 = m0 + ((lane >> 4) << 3);
#pragma unroll
  for (int j = 0; j < 8; ++j) {
    int r = row0 + j;
    if (r < nrows) C[(size_t)r * ldc + col] = c[j] + bv;
  }
}

// ---------------------------------------------------------------------------
// One wave per edge. The edge id is wave-uniform by construction; force it
// through readfirstlane so src/dst/dinv become scalar (SMEM) loads and the
// vector pipe only carries the float4 gather + 4 f32 atomics per lane.
__global__ void k_scatter(const long long* __restrict__ src,
                          const long long* __restrict__ dst,
                          const float* __restrict__ dinv,
                          const float* __restrict__ hw,
                          float* __restrict__ agg, int nedges) {
  int wid = blockIdx.x * (blockDim.x >> 5) + (threadIdx.x >> 5);
  int e = __builtin_amdgcn_readfirstlane(wid);   // wave-uniform edge id
  if (e >= nedges) return;
  int c = (threadIdx.x & 31) << 2;               // 4 floats per lane

  int s = __builtin_amdgcn_readfirstlane((int)src[e]);
  int d = __builtin_amdgcn_readfirstlane((int)dst[e]);
  float norm = dinv[s] * dinv[d];                // scalar loads, scalar mul

  const float4 v = *(const float4*)(hw + (size_t)s * DIM + c);
  float* out = agg + (size_t)d * DIM + c;
  __hip_atomic_fetch_add(out + 0, v.x * norm, __ATOMIC_RELAXED, __HIP_MEMORY_SCOPE_AGENT);
  __hip_atomic_fetch_add(out + 1, v.y * norm, __ATOMIC_RELAXED, __HIP_MEMORY_SCOPE_AGENT);
  __hip_atomic_fetch_add(out + 2, v.z * norm, __ATOMIC_RELAXED, __HIP_MEMORY_SCOPE_AGENT);
  __hip_atomic_fetch_add(out + 3, v.w * norm, __ATOMIC_RELAXED, __HIP_MEMORY_SCOPE_AGENT);
}

// ---------------------------------------------------------------------------
// h_{l+1} = elu(agg + dinv^2*hw + b) + h_l ; write into next column block.
__global__ void k_post(const float* __restrict__ agg, const float* __restrict__ hw,
                       const float* __restrict__ dinv, const float* __restrict__ b,
                       float* __restrict__ hcat, int l, int n, int ldh) {
  int g = blockIdx.x * blockDim.x + threadIdx.x;
  if (g >= n * DIM) return;
  int i = g >> 7, c = g & (DIM - 1);
  float di = dinv[i];
  float v = agg[g] + di * di * hw[g] + b[c];
  v = (v > 0.0f) ? v : (expf(v) - 1.0f);   // ELU(alpha=1)
  size_t base = (size_t)i * ldh;
  float res = hcat[base + (size_t)l * DIM + c];
  hcat[base + (size_t)(l + 1) * DIM + c] = v + res;
}

// ---------------------------------------------------------------------------
extern "C" void kernel_launch(void* const* d_in, const int* in_sizes, int n_in,
                              void* d_out, int out_size, void* d_ws, size_t ws_size,
                              hipStream_t stream) {
  const float*      x      = (const float*)d_in[0];
  const long long*  ei     = (const long long*)d_in[1];
  const float*      conv_w = (const float*)d_in[2];
  const float*      conv_b = (const float*)d_in[3];
  const float*      lin_w  = (const float*)d_in[4];
  const float*      lin_b  = (const float*)d_in[5];

  const int n  = in_sizes[0] / DIM;          // 100000
  const int e  = in_sizes[1] / 2;            // 1600000
  const int L  = in_sizes[2] / (DIM * DIM);  // 3
  const int DC = DIM * (L + 1);              // 512

  float* ws   = (float*)d_ws;
  float* hcat = ws;                               // n * DC
  float* hw   = hcat + (size_t)n * DC;            // n * DIM
  float* agg  = hw   + (size_t)n * DIM;           // n * DIM
  float* deg  = agg  + (size_t)n * DIM;           // n
  float* dinv = deg  + n;                         // n

  const long long* srcp = ei;
  const long long* dstp = ei + e;

  const int T = 256;
  const int gND = (n * DIM + T - 1) / T;
  const int gN  = (n + T - 1) / T;
  const int gE  = (e + T - 1) / T;
  const int gMM = (n + 15) / 16;                  // row tiles (6250)
  const int wavesPerBlock = T >> 5;               // 8
  const int gSC = (e + wavesPerBlock - 1) / wavesPerBlock;

  k_copy_x  <<<gND, T, 0, stream>>>(x, hcat, n, DC);
  k_deg_init<<<gN,  T, 0, stream>>>(deg, n);
  k_deg_edges<<<gE, T, 0, stream>>>(dstp, deg, e);
  k_dinv    <<<gN,  T, 0, stream>>>(deg, dinv, n);

  for (int l = 0; l < L; ++l) {
    hipMemsetAsync(agg, 0, (size_t)n * DIM * sizeof(float), stream);
    k_gemm_wmma<<<gMM, T, 0, stream>>>(hcat + (size_t)l * DIM, DC,
                                       conv_w + (size_t)l * DIM * DIM, DIM,
                                       nullptr, hw, DIM, DIM, n);
    k_scatter<<<gSC, T, 0, stream>>>(srcp, dstp, dinv, hw, agg, e);
    k_post<<<gND, T, 0, stream>>>(agg, hw, dinv, conv_b + (size_t)l * DIM,
                                  hcat, l, n, DC);
  }

  // out = hcat @ lin_w^T + lin_b   (K = 512)
  k_gemm_wmma<<<gMM, T, 0, stream>>>(hcat, DC, lin_w, DC, lin_b,
                                     (float*)d_out, DIM, DC, n);
}